// GroupDRO_36799279792334
// MI455X (gfx1250) — compile-verified
//
#include <hip/hip_runtime.h>
#include <stdint.h>

// GroupDRO on MI455X (gfx1250, wave32).
// Memory-bound: 128 MiB of one-shot input @ 23.3 TB/s => ~5.7us floor.
// - Kernel 1: per-block FULL 10,000-bin histogram in LDS (80 KB/block; only
//   possible with CDNA5's 320 KB-per-WGP LDS). One packed ds_add_u64 per
//   sample: count in bits [63:40], Q20 fixed-point loss sum in bits [39:0].
//   Input streamed with NON-TEMPORAL b128 loads (CDNA5 TH=NT) so the 192 MB
//   L2 keeps the 20 MB of partials resident for the next pass. Blocks flush
//   partials with plain coalesced stores (no contended global atomics).
// - Kernel 2a: 2D grid (groups x slot-chunks) collapses the P partial slots
//   into per-group f32 (sum, count) with lightly-contended global f32 atomics.
// - Kernel 2b: per-group exp-weighting + block reduction into 2 scalars.
// - Kernel 3: scalar divide.

#define MASK40 ((1ULL << 40) - 1ULL)
#define Q20 1048576.0f   // 2^20 fixed-point scale; losses are in [0,1)

typedef float v4f __attribute__((ext_vector_type(4)));
typedef int   v4i __attribute__((ext_vector_type(4)));

__device__ __forceinline__ unsigned long long pack_sample(float l) {
    return (1ULL << 40) | (unsigned long long)__float2uint_rn(l * Q20);
}

// ---------------------------------------------------------------- kernel 1 --
__global__ void groupdro_hist_kernel(const float* __restrict__ losses,
                                     const int*   __restrict__ gids,
                                     unsigned long long* __restrict__ partials,
                                     float* __restrict__ gsum2,  // 2*G floats (gsum|gcnt)
                                     float* __restrict__ acc,
                                     int n, int G) {
    extern __shared__ unsigned long long hist[];  // G packed u64 bins (80 KB)
    const int tid = threadIdx.x;
    const int nth = blockDim.x;

    for (int j = tid; j < G; j += nth) hist[j] = 0ULL;
    // Block 0 also zero-inits the f32 accumulation arrays for kernel 2a/2b.
    if (blockIdx.x == 0) {
        for (int j = tid; j < 2 * G; j += nth) gsum2[j] = 0.0f;
        if (tid == 0) { acc[0] = 0.0f; acc[1] = 0.0f; }
    }
    __syncthreads();

    // Non-temporal vectorized grid-stride stream, 2x unrolled: four
    // global_load_b128 (TH=NT) in flight per wave per iteration.
    const int nv = n >> 2;
    const v4f* __restrict__ lv = (const v4f*)losses;
    const v4i* __restrict__ iv = (const v4i*)gids;
    const int stride = gridDim.x * nth;
    for (int i = blockIdx.x * nth + tid; i < nv; i += 2 * stride) {
        const v4f l0 = __builtin_nontemporal_load(lv + i);
        const v4i g0 = __builtin_nontemporal_load(iv + i);
        const int i1 = i + stride;
        v4f l1; v4i g1;
        const bool has1 = (i1 < nv);
        if (has1) {
            l1 = __builtin_nontemporal_load(lv + i1);
            g1 = __builtin_nontemporal_load(iv + i1);
        }
        atomicAdd(&hist[g0.x], pack_sample(l0.x));
        atomicAdd(&hist[g0.y], pack_sample(l0.y));
        atomicAdd(&hist[g0.z], pack_sample(l0.z));
        atomicAdd(&hist[g0.w], pack_sample(l0.w));
        if (has1) {
            atomicAdd(&hist[g1.x], pack_sample(l1.x));
            atomicAdd(&hist[g1.y], pack_sample(l1.y));
            atomicAdd(&hist[g1.z], pack_sample(l1.z));
            atomicAdd(&hist[g1.w], pack_sample(l1.w));
        }
    }
    // Tail (n % 4), handled by block 0.
    if (blockIdx.x == 0) {
        const int base = nv << 2;
        if (tid < (n - base)) {
            atomicAdd(&hist[gids[base + tid]], pack_sample(losses[base + tid]));
        }
    }
    __syncthreads();

    // Private flush: coalesced b64 stores (default RT => stays in L2).
    unsigned long long* __restrict__ slot = partials + (size_t)blockIdx.x * (size_t)G;
    for (int j = tid; j < G; j += nth) slot[j] = hist[j];
}

// --------------------------------------------------------------- kernel 2a --
// Collapse P partial slots into per-group f32 sum/count. gridDim.y chunks the
// slot dimension so thousands of waves stream the (L2-resident) partials.
__global__ void groupdro_collapse_kernel(const unsigned long long* __restrict__ partials,
                                         float* __restrict__ gsum,
                                         float* __restrict__ gcnt,
                                         int G, int P) {
    const int g = blockIdx.x * blockDim.x + threadIdx.x;
    if (g >= G) return;
    const int p0 = blockIdx.y * 32;
    const int p1 = (p0 + 32 < P) ? (p0 + 32) : P;
    unsigned long long sQ = 0ULL;
    unsigned int cnt = 0u;
    for (int p = p0; p < p1; ++p) {   // coalesced across lanes for each p
        const unsigned long long v = partials[(size_t)p * (size_t)G + g];
        cnt += (unsigned int)(v >> 40);
        sQ  += (v & MASK40);
    }
    if (sQ | cnt) {
        atomicAdd(&gsum[g], (float)sQ * (1.0f / Q20));
        atomicAdd(&gcnt[g], (float)cnt);
    }
}

// --------------------------------------------------------------- kernel 2b --
__global__ void groupdro_weight_kernel(const float* __restrict__ gsum,
                                       const float* __restrict__ gcnt,
                                       const float* __restrict__ gw,
                                       float* __restrict__ acc,
                                       int G) {
    const int g = blockIdx.x * blockDim.x + threadIdx.x;
    float e = 0.0f, el = 0.0f;
    if (g < G) {
        const float c = gcnt[g];
        const float mean = (c > 0.0f) ? (gsum[g] / c) : 0.0f;
        e  = gw[g] * expf(0.01f * mean);   // STEP_SIZE = 0.01
        el = e * mean;
    }
    __shared__ float se[256], sel[256];
    const int tid = threadIdx.x;
    se[tid] = e; sel[tid] = el;
    __syncthreads();
    for (int s = 128; s > 0; s >>= 1) {
        if (tid < s) { se[tid] += se[tid + s]; sel[tid] += sel[tid + s]; }
        __syncthreads();
    }
    if (tid == 0) {
        atomicAdd(&acc[0], se[0]);    // sum(w)
        atomicAdd(&acc[1], sel[0]);   // sum(w * L)
    }
}

// ---------------------------------------------------------------- kernel 3 --
__global__ void groupdro_final_kernel(const float* __restrict__ acc,
                                      float* __restrict__ out) {
    if (threadIdx.x == 0) out[0] = acc[1] / acc[0];
}

// ------------------------------------------------------------------ launch --
extern "C" void kernel_launch(void* const* d_in, const int* in_sizes, int n_in,
                              void* d_out, int out_size, void* d_ws, size_t ws_size,
                              hipStream_t stream) {
    const float* losses = (const float*)d_in[0];
    const int*   gids   = (const int*)  d_in[1];
    const float* gw     = (const float*)d_in[2];
    const int n = in_sizes[0];
    const int G = in_sizes[2];
    float* out = (float*)d_out;

    // Workspace layout:
    //   [0, 1024)                : 2 f32 scalar accumulators (padded)
    //   [1024, 1024+4G)          : gsum  (f32 per group)
    //   [1024+4G, 1024+8G)       : gcnt  (f32 per group)
    //   [1024+8G, ...)           : P slots of G packed u64 partials
    char* base = (char*)d_ws;
    float* acc  = (float*)base;
    float* gsum = (float*)(base + 1024);
    float* gcnt = (float*)(base + 1024 + (size_t)4 * G);
    unsigned long long* partials =
        (unsigned long long*)(base + 1024 + (size_t)8 * G);

    const size_t headBytes = 1024 + (size_t)8 * G;
    const size_t slotBytes = (size_t)G * 8u;
    const size_t avail = (ws_size > headBytes) ? (ws_size - headBytes) : 0;
    long long Pll = (slotBytes > 0) ? (long long)(avail / slotBytes) : 1;
    if (Pll < 1)   Pll = 1;      // assume harness scratch >= ~1 MB
    if (Pll > 256) Pll = 256;    // 256 blocks x 32 waves => ~16 MB loads in flight
    const int P = (int)Pll;

    const int threads1 = 1024;                    // 32 wave32 waves / block
    const size_t ldsBytes = (size_t)G * 8u;       // 80 KB dynamic LDS
    hipLaunchKernelGGL(groupdro_hist_kernel, dim3(P), dim3(threads1), ldsBytes,
                       stream, losses, gids, partials, gsum, acc, n, G);

    const int t2 = 256;
    const int gx = (G + t2 - 1) / t2;
    const int gy = (P + 31) / 32;
    hipLaunchKernelGGL(groupdro_collapse_kernel, dim3(gx, gy), dim3(t2), 0,
                       stream, partials, gsum, gcnt, G, P);

    hipLaunchKernelGGL(groupdro_weight_kernel, dim3(gx), dim3(t2), 0,
                       stream, gsum, gcnt, gw, acc, G);

    hipLaunchKernelGGL(groupdro_final_kernel, dim3(1), dim3(32), 0,
                       stream, acc, out);
}